// NWGformerT_60198261620974
// MI455X (gfx1250) — compile-verified
//
#include <hip/hip_runtime.h>
#include <hip/hip_bf16.h>

typedef float v2f __attribute__((ext_vector_type(2)));
typedef float v8f __attribute__((ext_vector_type(8)));

#define PI_F 3.14159265358979323846f

// ---- order-preserving float<->uint encoding for atomic min/max ----
__device__ __forceinline__ unsigned fenc(float f) {
    unsigned u = __float_as_uint(f);
    return (u & 0x80000000u) ? ~u : (u | 0x80000000u);
}
__device__ __forceinline__ float fdec(unsigned u) {
    return __uint_as_float((u & 0x80000000u) ? (u & 0x7fffffffu) : ~u);
}
__device__ __forceinline__ float wred(float v) {
    #pragma unroll
    for (int m = 16; m >= 1; m >>= 1) v += __shfl_xor(v, m, 32);
    return v;
}
__device__ __forceinline__ unsigned umn(unsigned a, unsigned b){ return a < b ? a : b; }
__device__ __forceinline__ unsigned umx(unsigned a, unsigned b){ return a > b ? a : b; }

// ---------------------------------------------------------------------------
// 0) init: zero G(=out), rowsum, M, sums; seed min/max cells
// mm layout: [0]=xhmin [1]=xhmax [2]=smin [3]=smax [4]=hmin [5]=hmax
// ---------------------------------------------------------------------------
__global__ __launch_bounds__(256) void k_init(float* __restrict__ out,
                                              float* __restrict__ rowsum,
                                              float* __restrict__ M,
                                              float* __restrict__ sums,
                                              unsigned* __restrict__ mm, int n) {
    long long total = (long long)n * 64;
    long long stride = (long long)gridDim.x * 256;
    for (long long i = (long long)blockIdx.x * 256 + threadIdx.x; i < total; i += stride)
        out[i] = 0.f;
    int idx = blockIdx.x * 256 + threadIdx.x;
    for (int i = idx; i < n; i += gridDim.x * 256) rowsum[i] = 0.f;
    for (int i = idx; i < 4 * 4096; i += gridDim.x * 256) M[i] = 0.f;
    for (int i = idx; i < 256; i += gridDim.x * 256) sums[i] = 0.f;
    if (idx < 6) mm[idx] = (idx & 1) ? 0u : 0xFFFFFFFFu;
}

// ---------------------------------------------------------------------------
// 1) fused 4-way projection GEMM with V_WMMA_F32_16X16X4_F32
//    Q=sigmoid(x@Wq+bq), Vv=x@Wv+bv, Vg=x@Wg+bg, Qg=relu(x@Wg2+bg2)
//    block = 512 threads = 16 waves = 4 matrices x 4 col-tiles; 16 rows/block
// ---------------------------------------------------------------------------
__global__ __launch_bounds__(512) void k_gemm4(
    const float* __restrict__ x,
    const float* __restrict__ Wq, const float* __restrict__ bq,
    const float* __restrict__ Wv, const float* __restrict__ bv,
    const float* __restrict__ Wg, const float* __restrict__ bg,
    const float* __restrict__ Wg2, const float* __restrict__ bg2,
    float* __restrict__ Qb, float* __restrict__ Vvb,
    float* __restrict__ Vgb, float* __restrict__ Qgb, int n) {
    __shared__ float xs[16 * 132];
    const int r0 = blockIdx.x * 16;
    const int tid = threadIdx.x;
    #pragma unroll
    for (int t = 0; t < 4; t++) {
        int lin = tid + t * 512;
        int rr = lin >> 7, cc = lin & 127;
        int grow = r0 + rr;
        xs[rr * 132 + cc] = (grow < n) ? x[(size_t)grow * 128 + cc] : 0.f;
    }
    __syncthreads();

    const int w = tid >> 5, lane = tid & 31;
    const int m = w >> 2, ct = w & 3;
    const int half = lane >> 4, lm = lane & 15;
    const float* W = (m == 0) ? Wq : (m == 1) ? Wv : (m == 2) ? Wg : Wg2;
    const float* B = (m == 0) ? bq : (m == 1) ? bv : (m == 2) ? bg : bg2;
    float* O       = (m == 0) ? Qb : (m == 1) ? Vvb : (m == 2) ? Vgb : Qgb;
    const int col = ct * 16 + lm;

    v8f acc = {};
    #pragma unroll 4
    for (int kk = 0; kk < 32; ++kk) {
        int k0 = kk * 4 + 2 * half;
        v2f a, b;
        a.x = xs[lm * 132 + k0];
        a.y = xs[lm * 132 + k0 + 1];
        b.x = W[k0 * 64 + col];
        b.y = W[(k0 + 1) * 64 + col];
        acc = __builtin_amdgcn_wmma_f32_16x16x4_f32(
            false, a, false, b, (short)0, acc, false, false);
    }

    float bb = B[col];
    #pragma unroll
    for (int r = 0; r < 8; r++) {
        int grow = r0 + r + 8 * half;
        if (grow >= n) continue;
        float v = acc[r] + bb;
        if (m == 0) v = 1.f / (1.f + expf(-v));
        else if (m == 3) v = fmaxf(v, 0.f);
        O[(size_t)grow * 64 + col] = v;
    }
}

// ---------------------------------------------------------------------------
// 2) per-row layernorms: Qn = relu(ln(Q)) in place, xH = sum(Qn),
//    h = (sum ln(Qg))^2 ; block-reduced min/max of xH and S
// ---------------------------------------------------------------------------
__global__ __launch_bounds__(256) void k_rownorm(
    float* __restrict__ Qb, const float* __restrict__ Qgb,
    const float* __restrict__ S, float* __restrict__ xH, float* __restrict__ h,
    unsigned* __restrict__ mm, int n) {
    __shared__ unsigned sxmn[8], sxmx[8], ssmn[8], ssmx[8];
    const int wv = threadIdx.x >> 5, lane = threadIdx.x & 31;
    const int row = blockIdx.x * 8 + wv;
    float xhv = 0.f, sv = 0.f;
    const bool valid = (row < n);
    if (valid) {
        size_t base = (size_t)row * 64;
        float q0 = Qb[base + lane], q1 = Qb[base + lane + 32];
        float mean = wred(q0 + q1) * (1.f / 64.f);
        float d0 = q0 - mean, d1 = q1 - mean;
        float inv = rsqrtf(wred(d0 * d0 + d1 * d1) * (1.f / 64.f) + 1e-5f);
        float n0 = fmaxf(d0 * inv, 0.f), n1 = fmaxf(d1 * inv, 0.f);
        Qb[base + lane] = n0;
        Qb[base + lane + 32] = n1;
        xhv = wred(n0 + n1);                 // P_T = 1
        if (lane == 0) xH[row] = xhv;

        float g0 = Qgb[base + lane], g1 = Qgb[base + lane + 32];
        float gm = wred(g0 + g1) * (1.f / 64.f);
        float e0 = g0 - gm, e1 = g1 - gm;
        float gi = rsqrtf(wred(e0 * e0 + e1 * e1) * (1.f / 64.f) + 1e-5f);
        float ls = wred((e0 + e1) * gi);
        if (lane == 0) h[row] = ls * ls;     // P_GAT = 2
        sv = S[row];
    }
    if (lane == 0) {
        sxmn[wv] = valid ? fenc(xhv) : 0xFFFFFFFFu;
        sxmx[wv] = valid ? fenc(xhv) : 0u;
        ssmn[wv] = valid ? fenc(sv)  : 0xFFFFFFFFu;
        ssmx[wv] = valid ? fenc(sv)  : 0u;
    }
    __syncthreads();
    if (threadIdx.x == 0) {
        unsigned a = 0xFFFFFFFFu, b = 0u, c = 0xFFFFFFFFu, d = 0u;
        #pragma unroll
        for (int i = 0; i < 8; i++) {
            a = umn(a, sxmn[i]); b = umx(b, sxmx[i]);
            c = umn(c, ssmn[i]); d = umx(d, ssmx[i]);
        }
        atomicMin(&mm[0], a); atomicMax(&mm[1], b);
        atomicMin(&mm[2], c); atomicMax(&mm[3], d);
    }
}

// ---------------------------------------------------------------------------
// 3) min/max of h[src] across edges
// ---------------------------------------------------------------------------
__global__ __launch_bounds__(256) void k_edge_mm(const int* __restrict__ ei,
                                                 const float* __restrict__ h,
                                                 unsigned* __restrict__ mm, int E) {
    __shared__ unsigned smn[8], smx[8];
    unsigned mn = 0xFFFFFFFFu, mx = 0u;
    for (int e = blockIdx.x * 256 + threadIdx.x; e < E; e += gridDim.x * 256) {
        unsigned u = fenc(h[ei[e]]);
        mn = umn(mn, u); mx = umx(mx, u);
    }
    #pragma unroll
    for (int m = 16; m >= 1; m >>= 1) {
        mn = umn(mn, (unsigned)__shfl_xor((int)mn, m, 32));
        mx = umx(mx, (unsigned)__shfl_xor((int)mx, m, 32));
    }
    int wv = threadIdx.x >> 5, lane = threadIdx.x & 31;
    if (lane == 0) { smn[wv] = mn; smx[wv] = mx; }
    __syncthreads();
    if (threadIdx.x == 0) {
        unsigned a = 0xFFFFFFFFu, b = 0u;
        #pragma unroll
        for (int i = 0; i < 8; i++) { a = umn(a, smn[i]); b = umx(b, smx[i]); }
        atomicMin(&mm[4], a); atomicMax(&mm[5], b);
    }
}

// ---------------------------------------------------------------------------
// 4) phase coefficients: sc[0]=cx, sc[1]=cs, sc[2]=cg
// ---------------------------------------------------------------------------
__global__ void k_scalars(const unsigned* __restrict__ mm, float* __restrict__ sc) {
    if (threadIdx.x == 0 && blockIdx.x == 0) {
        float mxr = (fdec(mm[1]) - fdec(mm[0])) * 2.0f;  // M_T
        float msr = (fdec(mm[3]) - fdec(mm[2])) * 2.0f;  // M_T
        float mgr = (fdec(mm[5]) - fdec(mm[4])) * 1.0f;  // M_GAT
        sc[0] = PI_F / (2.f * mxr + 1e-10f);
        sc[1] = PI_F / (2.f * msr + 1e-10f);
        sc[2] = PI_F / (2.f * mgr + 1e-10f);
    }
}

// ---------------------------------------------------------------------------
// 5) per-node phase weights: wcx=cos(cx*xH), wsx=sin(cx*xH),
//                            wcs=cos(cs*S),  wss=sin(cs*S)
// ---------------------------------------------------------------------------
__global__ __launch_bounds__(256) void k_phase(
    const float* __restrict__ xH, const float* __restrict__ S,
    const float* __restrict__ sc, float* __restrict__ wcx, float* __restrict__ wsx,
    float* __restrict__ wcs, float* __restrict__ wss, int n) {
    int i = blockIdx.x * 256 + threadIdx.x;
    if (i >= n) return;
    float phx = sc[0] * xH[i], phs = sc[1] * S[i];
    wcx[i] = cosf(phx); wsx[i] = sinf(phx);
    wcs[i] = cosf(phs); wss[i] = sinf(phs);
}

// ---------------------------------------------------------------------------
// 6) WMMA column reductions over K=N:
//    M[m][i][j] = sum_k w_k(m)*Qn[k][i]*Vv[k][j], sums[m][i] = sum_k w_k(m)*Qn[k][i]
//    grid = (chunks, 4 matrices); block = 512 = 16 waves = 4x4 (i-tile, j-tile)
// ---------------------------------------------------------------------------
__global__ __launch_bounds__(512) void k_colreduce(
    const float* __restrict__ Qn, const float* __restrict__ Vv,
    const float* __restrict__ wcx, const float* __restrict__ wsx,
    const float* __restrict__ wcs, const float* __restrict__ wss,
    float* __restrict__ M, float* __restrict__ sums, int n) {
    const int m = blockIdx.y;
    const float* wp = (m == 0) ? wcx : (m == 1) ? wsx : (m == 2) ? wcs : wss;
    const int tid = threadIdx.x;
    const int w = tid >> 5, lane = tid & 31;
    const int it = w >> 2, jt = w & 3;
    const int half = lane >> 4, lm = lane & 15;
    const int ia = it * 16 + lm, ja = jt * 16 + lm;

    int rpb = (((n + gridDim.x - 1) / gridDim.x) + 3) & ~3;
    int kbeg = blockIdx.x * rpb;
    int kend = min(n, kbeg + rpb);

    v8f acc = {};
    float sacc = 0.f;
    for (int k0 = kbeg; k0 < kend; k0 += 4) {
        int k = k0 + 2 * half;
        bool v0 = k < kend, v1 = (k + 1) < kend;
        size_t k64 = (size_t)k * 64;
        v2f a, b;
        // A = cq^T: A[i][k] = w_k * Qn[k][i]  (16x4 f32 fragment)
        a.x = v0 ? wp[k]     * Qn[k64 + ia]      : 0.f;
        a.y = v1 ? wp[k + 1] * Qn[k64 + 64 + ia] : 0.f;
        // B[k][j] = Vv[k][j] (4x16 f32 fragment)
        b.x = v0 ? Vv[k64 + ja]      : 0.f;
        b.y = v1 ? Vv[k64 + 64 + ja] : 0.f;
        acc = __builtin_amdgcn_wmma_f32_16x16x4_f32(
            false, a, false, b, (short)0, acc, false, false);
        if (jt == 0) sacc += a.x + a.y;   // column sums of cq for free
    }
    #pragma unroll
    for (int r = 0; r < 8; r++)
        atomicAdd(&M[m * 4096 + (it * 16 + r + 8 * half) * 64 + jt * 16 + lm], acc[r]);
    if (jt == 0) atomicAdd(&sums[m * 64 + ia], sacc);
}

// ---------------------------------------------------------------------------
// 7) per-edge attention + rowsum
// ---------------------------------------------------------------------------
__global__ __launch_bounds__(256) void k_edge_att(
    const int* __restrict__ ei, const float* __restrict__ h,
    const float* __restrict__ sc, float* __restrict__ att,
    float* __restrict__ rowsum, int E) {
    int e = blockIdx.x * 256 + threadIdx.x;
    if (e >= E) return;
    int s = ei[e], d = ei[E + e];
    float a = cosf(sc[2] * (h[s] - h[d]));
    att[e] = a;
    atomicAdd(&rowsum[s], a);
}

// ---------------------------------------------------------------------------
// 8) edge message scatter: G[src] += (att/rowsum[src] + eps) * Vg[dst]
// ---------------------------------------------------------------------------
__global__ __launch_bounds__(256) void k_edge_msg(
    const int* __restrict__ ei, const float* __restrict__ att,
    const float* __restrict__ rowsum, const float* __restrict__ Vg,
    float* __restrict__ G, int E) {
    int wv = threadIdx.x >> 5, lane = threadIdx.x & 31;
    int e = blockIdx.x * 8 + wv;
    if (e >= E) return;
    int s = ei[e], d = ei[E + e];
    float natt = att[e] / rowsum[s] + 1e-10f;
    size_t db = (size_t)d * 64, sb = (size_t)s * 64;
    atomicAdd(&G[sb + lane],      natt * Vg[db + lane]);
    atomicAdd(&G[sb + lane + 32], natt * Vg[db + lane + 32]);
}

// ---------------------------------------------------------------------------
// 9) WMMA finalize: T_m = Qn @ M_m (N x 64 x 64 GEMMs), then
//    out = relu((0.8*(cx*T1+sx*T2)+0.2*(cs*T3+ss*T4))/den + gat)
//    block = 256 = 8 waves = (2 row-tiles x 4 col-tiles), 32 rows/block
// ---------------------------------------------------------------------------
__global__ __launch_bounds__(256) void k_former_final(
    const float* __restrict__ Qn,
    const float* __restrict__ wcx, const float* __restrict__ wsx,
    const float* __restrict__ wcs, const float* __restrict__ wss,
    const float* __restrict__ M, const float* __restrict__ sums,
    float* __restrict__ out, int n) {
    __shared__ float qs[32 * 68];
    __shared__ float rcx[32], rsx[32], rcs[32], rss[32], rid[32];
    const int base = blockIdx.x * 32;
    const int tid = threadIdx.x;
    #pragma unroll
    for (int t = 0; t < 8; t++) {
        int lin = tid + t * 256;
        int rr = lin >> 6, cc = lin & 63;
        int grow = base + rr;
        qs[rr * 68 + cc] = (grow < n) ? Qn[(size_t)grow * 64 + cc] : 0.f;
    }
    __syncthreads();
    if (tid < 32) {
        int grow = base + tid;
        float cxv = 0, sxv = 0, csv = 0, ssv = 0, ivd = 0;
        if (grow < n) {
            cxv = wcx[grow]; sxv = wsx[grow]; csv = wcs[grow]; ssv = wss[grow];
            float d1 = 0, d2 = 0, d3 = 0, d4 = 0;
            for (int i = 0; i < 64; i++) {
                float q = qs[tid * 68 + i];
                d1 = fmaf(q, sums[i],       d1);
                d2 = fmaf(q, sums[64 + i],  d2);
                d3 = fmaf(q, sums[128 + i], d3);
                d4 = fmaf(q, sums[192 + i], d4);
            }
            float den = 0.8f * (cxv * d1 + sxv * d2) + 0.2f * (csv * d3 + ssv * d4) + 1e-10f;
            ivd = 1.f / den;
        }
        rcx[tid] = cxv; rsx[tid] = sxv; rcs[tid] = csv; rss[tid] = ssv; rid[tid] = ivd;
    }
    __syncthreads();

    const int w = tid >> 5, lane = tid & 31;
    const int rt = w >> 2, ct = w & 3;
    const int half = lane >> 4, lm = lane & 15;
    const int col = ct * 16 + lm;
    v8f a1 = {}, a2 = {}, a3 = {}, a4 = {};
    #pragma unroll 2
    for (int km = 0; km < 16; ++km) {
        int k0 = km * 4 + 2 * half;
        v2f a, b;
        a.x = qs[(rt * 16 + lm) * 68 + k0];
        a.y = qs[(rt * 16 + lm) * 68 + k0 + 1];
        b.x = M[k0 * 64 + col];            b.y = M[(k0 + 1) * 64 + col];
        a1 = __builtin_amdgcn_wmma_f32_16x16x4_f32(false, a, false, b, (short)0, a1, false, false);
        b.x = M[4096 + k0 * 64 + col];     b.y = M[4096 + (k0 + 1) * 64 + col];
        a2 = __builtin_amdgcn_wmma_f32_16x16x4_f32(false, a, false, b, (short)0, a2, false, false);
        b.x = M[2 * 4096 + k0 * 64 + col]; b.y = M[2 * 4096 + (k0 + 1) * 64 + col];
        a3 = __builtin_amdgcn_wmma_f32_16x16x4_f32(false, a, false, b, (short)0, a3, false, false);
        b.x = M[3 * 4096 + k0 * 64 + col]; b.y = M[3 * 4096 + (k0 + 1) * 64 + col];
        a4 = __builtin_amdgcn_wmma_f32_16x16x4_f32(false, a, false, b, (short)0, a4, false, false);
    }
    #pragma unroll
    for (int r = 0; r < 8; r++) {
        int rl = rt * 16 + r + 8 * half;
        int grow = base + rl;
        if (grow >= n) continue;
        float num = 0.8f * (rcx[rl] * a1[r] + rsx[rl] * a2[r]) +
                    0.2f * (rcs[rl] * a3[r] + rss[rl] * a4[r]);
        size_t o = (size_t)grow * 64 + col;
        float v = num * rid[rl] + out[o];
        out[o] = fmaxf(v, 0.f);
    }
}

// ---------------------------------------------------------------------------
extern "C" void kernel_launch(void* const* d_in, const int* in_sizes, int n_in,
                              void* d_out, int out_size, void* d_ws, size_t ws_size,
                              hipStream_t stream) {
    const float* x   = (const float*)d_in[0];
    const int*   ei  = (const int*)d_in[1];
    const float* S   = (const float*)d_in[2];
    const float* Wq  = (const float*)d_in[3];
    const float* bq  = (const float*)d_in[4];
    const float* Wv  = (const float*)d_in[5];
    const float* bv  = (const float*)d_in[6];
    const float* Wg  = (const float*)d_in[7];
    const float* bg  = (const float*)d_in[8];
    const float* Wg2 = (const float*)d_in[9];
    const float* bg2 = (const float*)d_in[10];
    const int N = in_sizes[0] / 128;
    const int E = in_sizes[1] / 2;

    float* ws   = (float*)d_ws;
    float* Qn   = ws;                          // N*64 (sigmoid(Q) -> Qn in place)
    float* Vv   = ws + (size_t)N * 64;         // N*64
    float* Vg   = ws + (size_t)N * 64 * 2;     // N*64
    float* Qg   = ws + (size_t)N * 64 * 3;     // N*64 (relu Qg); reused as att[E]
    float* att  = Qg;
    float* xH   = ws + (size_t)N * 64 * 4;     // N
    float* h    = xH + N;                      // N
    float* rsm  = h + N;                       // N
    float* wcx  = rsm + N;                     // N
    float* wsx  = wcx + N;                     // N
    float* wcs  = wsx + N;                     // N
    float* wss  = wcs + N;                     // N
    float* M    = wss + N;                     // 4*4096
    float* sums = M + 4 * 4096;                // 4*64
    float* sc   = sums + 256;                  // 4
    unsigned* mm = (unsigned*)(sc + 4);        // 6
    float* out = (float*)d_out;

    k_init<<<1024, 256, 0, stream>>>(out, rsm, M, sums, mm, N);
    k_gemm4<<<(N + 15) / 16, 512, 0, stream>>>(x, Wq, bq, Wv, bv, Wg, bg, Wg2, bg2,
                                               Qn, Vv, Vg, Qg, N);
    k_rownorm<<<(N + 7) / 8, 256, 0, stream>>>(Qn, Qg, S, xH, h, mm, N);
    k_edge_mm<<<1024, 256, 0, stream>>>(ei, h, mm, E);
    k_scalars<<<1, 32, 0, stream>>>(mm, sc);
    k_phase<<<(N + 255) / 256, 256, 0, stream>>>(xH, S, sc, wcx, wsx, wcs, wss, N);
    k_colreduce<<<dim3(64, 4), 512, 0, stream>>>(Qn, Vv, wcx, wsx, wcs, wss, M, sums, N);
    k_edge_att<<<(E + 255) / 256, 256, 0, stream>>>(ei, h, sc, att, rsm, E);
    k_edge_msg<<<(E + 7) / 8, 256, 0, stream>>>(ei, att, rsm, Vg, out, E);
    k_former_final<<<(N + 31) / 32, 256, 0, stream>>>(Qn, wcx, wsx, wcs, wss, M, sums, out, N);
}